// Correlation_network_72155450572967
// MI455X (gfx1250) — compile-verified
//
#include <hip/hip_runtime.h>

typedef __attribute__((ext_vector_type(2))) float v2f;
typedef __attribute__((ext_vector_type(8))) float v8f;
typedef __attribute__((ext_vector_type(4))) unsigned int u32x4;
typedef __attribute__((ext_vector_type(8))) int i32x8;
typedef __attribute__((ext_vector_type(4))) int i32x4;

#define NB   8        // batches
#define MN   2304     // 48*48 positions
#define CK   256      // channels (K)
#define BM   128
#define BN   128
#define BK   32
#define NSTG (CK / BK)  // 8 K-stages
#define LSTR 36       // LDS row stride in floats (BK + 4 pad dwords)

#if __has_builtin(__builtin_amdgcn_tensor_load_to_lds)
#define USE_TDM 1
#else
#define USE_TDM 0
#endif

#if USE_TDM
// ---------------------------------------------------------------------------
// Issue one TDM descriptor: 2D tile (BK floats x BM rows) from a row-major
// [*, CK] f32 tensor into LDS, with hardware padding of 4 dwords every
// 32 dwords so the landed layout has row stride LSTR=36 (bank-conflict-free).
// D# bit packing per CDNA5 ISA ch.8 (group0 128b, group1 256b).
// 6-arg builtin form (clang-23 / therock-10.0 signature).
// ---------------------------------------------------------------------------
static __device__ __forceinline__ void tdm_issue(const float* gptr, unsigned lds_off) {
    unsigned long long ga = (unsigned long long)(size_t)gptr;
    u32x4 g0 = {0u, 0u, 0u, 0u};
    g0[0] = 1u;                                      // count=1 valid, is_restore=0, gather off
    g0[1] = lds_off;                                 // lds_addr (bytes)
    g0[2] = (unsigned)ga;                            // global_addr[31:0]
    g0[3] = (unsigned)((ga >> 32) & 0x01FFFFFFull)   // global_addr[56:32]
          | (2u << 30);                              // type=2 ("image")
    i32x8 g1 = {0, 0, 0, 0, 0, 0, 0, 0};
    g1[0] = (2 << 16)       // data_size = 4 bytes
          | (1 << 20)       // pad_enable
          | (4 << 22)       // pad_interval: every 2^(4+1)=32 dwords
          | (3 << 25);      // pad_amount: 3+1 = 4 dwords
    g1[1] = (int)((unsigned)(CK & 0xFFFF) << 16);    // tensor_dim0[15:0] = 256
    g1[2] = (int)((unsigned)(MN & 0xFFFF) << 16);    // tensor_dim0[31:16]=0 | tensor_dim1[15:0]=2304
    g1[3] = (int)((unsigned)BK << 16);               // tensor_dim1[31:16]=0 | tile_dim0=32
    g1[4] = BM;                                      // tile_dim1=128 | tile_dim2=0
    g1[5] = CK;                                      // tensor_dim0_stride[31:0] = 256
    i32x4 z4 = {0, 0, 0, 0};
    i32x8 z8 = {0, 0, 0, 0, 0, 0, 0, 0};
    __builtin_amdgcn_tensor_load_to_lds(g0, g1, z4, z4, z8, 0);
}
#endif

// ---------------------------------------------------------------------------
// Pass 1: batched NT GEMM corr = (A @ B^T) / C, fused column sum-of-squares.
// 8 waves (wave32) arranged 4(M) x 2(N); each wave owns eight 16x16
// V_WMMA_F32_16X16X4_F32 accumulators (32x64 region of a 128x128 tile).
// TDM double-buffered staging: stage i+1's DMA overlaps stage i's WMMAs.
// ---------------------------------------------------------------------------
__global__ __launch_bounds__(256)
void corr_gemm_kernel(const float* __restrict__ A, const float* __restrict__ B,
                      float* __restrict__ out, float* __restrict__ colss) {
    __shared__ float Alds[2][BM * LSTR];
    __shared__ float Blds[2][BN * LSTR];

    const int b    = blockIdx.z;
    const int m0   = blockIdx.y * BM;
    const int n0   = blockIdx.x * BN;
    const int tid  = threadIdx.x;
    const int lane = tid & 31;
    const int wave = tid >> 5;
    const int waveM = (wave & 3) * 32;
    const int waveN = (wave >> 2) * 64;
    const int half  = (lane >> 4) & 1;
    const int lrow  = lane & 15;

    const float* __restrict__ Ab = A + ((size_t)b * MN + m0) * CK;
    const float* __restrict__ Bb = B + ((size_t)b * MN + n0) * CK;

    v8f acc[2][4];
#pragma unroll
    for (int t = 0; t < 2; ++t)
#pragma unroll
        for (int s = 0; s < 4; ++s)
            acc[t][s] = (v8f){0.f, 0.f, 0.f, 0.f, 0.f, 0.f, 0.f, 0.f};

#if USE_TDM
    if (wave == 0) {
        tdm_issue(Ab, (unsigned)(size_t)(void*)&Alds[0][0]);
        tdm_issue(Bb, (unsigned)(size_t)(void*)&Blds[0][0]);
    }
#else
    const int lr = tid >> 3;
    const int lc = (tid & 7) * 4;
#endif

#pragma unroll
    for (int i = 0; i < NSTG; ++i) {
        const int cur = i & 1;
#if USE_TDM
        if (wave == 0) {
            if (i + 1 < NSTG) {
                tdm_issue(Ab + (i + 1) * BK, (unsigned)(size_t)(void*)&Alds[cur ^ 1][0]);
                tdm_issue(Bb + (i + 1) * BK, (unsigned)(size_t)(void*)&Blds[cur ^ 1][0]);
                __builtin_amdgcn_s_wait_tensorcnt(2);  // stage i landed (in-order)
            } else {
                __builtin_amdgcn_s_wait_tensorcnt(0);
            }
        }
        __syncthreads();   // LDS buf[cur] visible to all waves
#else
#pragma unroll
        for (int rr = 0; rr < 4; ++rr) {
            const int row = lr + rr * 32;
            const float4 av = *(const float4*)(Ab + (size_t)row * CK + i * BK + lc);
            const float4 bv = *(const float4*)(Bb + (size_t)row * CK + i * BK + lc);
            *(float4*)&Alds[cur][row * LSTR + lc] = av;
            *(float4*)&Blds[cur][row * LSTR + lc] = bv;
        }
        __syncthreads();
#endif

#pragma unroll
        for (int k4 = 0; k4 < BK; k4 += 4) {
            v2f afr[2], bfr[4];
#pragma unroll
            for (int t = 0; t < 2; ++t)
                afr[t] = *(const v2f*)&Alds[cur][(waveM + t * 16 + lrow) * LSTR + k4 + 2 * half];
#pragma unroll
            for (int s = 0; s < 4; ++s)
                bfr[s] = *(const v2f*)&Blds[cur][(waveN + s * 16 + lrow) * LSTR + k4 + 2 * half];
#pragma unroll
            for (int t = 0; t < 2; ++t)
#pragma unroll
                for (int s = 0; s < 4; ++s)
                    acc[t][s] = __builtin_amdgcn_wmma_f32_16x16x4_f32(
                        false, afr[t], false, bfr[s],
                        (short)0, acc[t][s], false, false);
        }
        __syncthreads();   // all reads of buf[cur] done before it is refilled
    }

    // Epilogue: scale by 1/C, write corr, accumulate column sum-of-squares.
    // C/D layout: VGPR r of lane l holds (row = r + 8*(l>=16), col = l%16),
    // so each lane's 8 regs share one column -> cheap column reduction.
    const float scale = 1.0f / (float)CK;
    float* __restrict__ outb = out + (size_t)b * MN * MN;

#pragma unroll
    for (int s = 0; s < 4; ++s) {
        const int col = n0 + waveN + s * 16 + lrow;
        float ss = 0.f;
#pragma unroll
        for (int t = 0; t < 2; ++t) {
#pragma unroll
            for (int r = 0; r < 8; ++r) {
                float v = acc[t][s][r] * scale;
                acc[t][s][r] = v;
                ss += v * v;
                const int row = m0 + waveM + t * 16 + r + 8 * half;
                outb[(size_t)row * MN + col] = v;
            }
        }
        ss += __shfl_xor(ss, 16, 32);   // lane l + lane l^16: full 32-row column sum
        if (half == 0)
            atomicAdd(&colss[b * MN + col], ss);
    }
}

// ---------------------------------------------------------------------------
// Workspace zeroing (graph-capture safe, deterministic each launch)
// ---------------------------------------------------------------------------
__global__ void zero_kernel(float* __restrict__ p, int n) {
    int i = blockIdx.x * blockDim.x + threadIdx.x;
    if (i < n) p[i] = 0.f;
}

// ---------------------------------------------------------------------------
// Pass 2: out[b, m, n] /= sqrt(colss[b, n]); float4 per thread.
// ---------------------------------------------------------------------------
__global__ __launch_bounds__(256)
void corr_norm_kernel(float* __restrict__ out, const float* __restrict__ colss) {
    const size_t total4 = (size_t)NB * MN * MN / 4;
    size_t i4 = (size_t)blockIdx.x * blockDim.x + threadIdx.x;
    if (i4 >= total4) return;
    const size_t idx   = i4 * 4;
    const size_t per_b = (size_t)MN * MN;
    const int b = (int)(idx / per_b);
    const int n = (int)(idx % MN);          // MN % 4 == 0 -> 4 elems share b, n..n+3

    float4 v = *(float4*)(out + idx);
    const float4 sq = *(const float4*)(colss + (size_t)b * MN + n);
    v.x /= sqrtf(sq.x);
    v.y /= sqrtf(sq.y);
    v.z /= sqrtf(sq.z);
    v.w /= sqrtf(sq.w);
    *(float4*)(out + idx) = v;
}

extern "C" void kernel_launch(void* const* d_in, const int* in_sizes, int n_in,
                              void* d_out, int out_size, void* d_ws, size_t ws_size,
                              hipStream_t stream) {
    const float* A = (const float*)d_in[0];   // feature_A [8,48,48,256] f32
    const float* B = (const float*)d_in[1];   // feature_B [8,48,48,256] f32
    float* out   = (float*)d_out;             // [8,48,48,48,48] f32
    float* colss = (float*)d_ws;              // [8,2304] f32 column sum-of-squares

    const int nss = NB * MN;
    zero_kernel<<<(nss + 255) / 256, 256, 0, stream>>>(colss, nss);

    dim3 grid(MN / BN, MN / BM, NB);          // 18 x 18 x 8
    corr_gemm_kernel<<<grid, 256, 0, stream>>>(A, B, out, colss);

    const size_t total4 = (size_t)NB * MN * MN / 4;
    const int nblk = (int)((total4 + 255) / 256);
    corr_norm_kernel<<<nblk, 256, 0, stream>>>(out, colss);
}